// UGCN_83202106458213
// MI455X (gfx1250) — compile-verified
//
#include <hip/hip_runtime.h>

// ---------------------------------------------------------------------------
// UGCN on MI455X (gfx1250): L2-resident scatter/gather + exact WMMA bias add.
// ---------------------------------------------------------------------------

typedef __attribute__((ext_vector_type(2))) float v2f;
typedef __attribute__((ext_vector_type(8))) float v8f;

#define UGCN_EMB 64

#if defined(__has_builtin)
#if __has_builtin(__builtin_amdgcn_wmma_f32_16x16x4_f32)
#define UGCN_HAVE_WMMA 1
#endif
#endif

// Zero the user_h accumulator (first N*64 floats of d_out) and set deg = 1.0
// (self loops). Runs every call so graph replay is deterministic.
__global__ void ugcn_init(float* __restrict__ h, float* __restrict__ deg, int N) {
  long long t = (long long)blockIdx.x * blockDim.x + threadIdx.x;
  long long total = (long long)N * UGCN_EMB;
  if (t < total) h[t] = 0.0f;
  if (t < N) deg[t] = 1.0f;
}

// deg[row[e]] += 1 for every edge (segment_sum of ones over `row`).
__global__ void ugcn_degree(const int* __restrict__ rows, float* __restrict__ deg, int E) {
  int e = blockIdx.x * blockDim.x + threadIdx.x;
  if (e < E) atomicAdd(&deg[rows[e]], 1.0f);
}

__global__ void ugcn_rsqrt(const float* __restrict__ deg, float* __restrict__ dinv, int N) {
  int i = blockIdx.x * blockDim.x + threadIdx.x;
  if (i < N) dinv[i] = __frsqrt_rn(deg[i]);
}

// Main scatter: 16 lanes per edge, each lane handles one float4 (4 atomics).
// Gather of user_weight[col] is a coalesced b128 load per lane; h accumulator
// and user_weight both live in the 192MB L2, so atomics stay on-chip.
__global__ void ugcn_scatter(const int* __restrict__ rows, const int* __restrict__ cols,
                             const float* __restrict__ uw, const float* __restrict__ dinv,
                             float* __restrict__ h, int E) {
  long long t = (long long)blockIdx.x * blockDim.x + threadIdx.x;
  int e = (int)(t >> 4);
  if (e >= E) return;
  int sub = (int)(t & 15);
  int r = rows[e];
  int c = cols[e];
  float nrm = dinv[r] * dinv[c];
  float4 v = ((const float4*)(uw + (size_t)c * UGCN_EMB))[sub];
  float* dst = h + (size_t)r * UGCN_EMB + (size_t)sub * 4;
  atomicAdd(dst + 0, v.x * nrm);
  atomicAdd(dst + 1, v.y * nrm);
  atomicAdd(dst + 2, v.z * nrm);
  atomicAdd(dst + 3, v.w * nrm);
}

// Finalize: one wave32 per 16-node x 16-col tile.
// C tile = h + self-loop term, laid out per the ISA 16x16 f32 C/D mapping
// (VGPR j : lanes 0-15 -> row j, lanes 16-31 -> row j+8; lane&15 -> column).
// WMMA computes D = A(ones col K=0) x B(row K=0 = bias) + C  ==  C + bias,
// exactly (1.0*bias is exact in f32). Then relu and store in place.
__global__ void ugcn_finalize_wmma(float* __restrict__ h, const float* __restrict__ uw,
                                   const float* __restrict__ dinv,
                                   const float* __restrict__ bias, int tiles_m) {
  int wid = (int)(((long long)blockIdx.x * blockDim.x + threadIdx.x) >> 5);
  int lane = (int)(threadIdx.x & 31);
  int total = tiles_m * (UGCN_EMB / 16);
  if (wid >= total) return;  // uniform per wave: surviving waves have EXEC all-ones
  int tm = wid >> 2;
  int tn = wid & 3;
  int col = tn * 16 + (lane & 15);
  int row0 = tm * 16 + ((lane >> 4) << 3);  // +8 for lanes 16-31

  v8f c;
#pragma unroll
  for (int j = 0; j < 8; ++j) {
    int row = row0 + j;
    float di = dinv[row];
    size_t idx = (size_t)row * UGCN_EMB + col;
    c[j] = h[idx] + uw[idx] * (di * di);  // fold in self-loop term
  }

#ifdef UGCN_HAVE_WMMA
  // A (16x4): VGPR0 = {K=0 lanes 0-15, K=2 lanes 16-31}; VGPR1 = {K=1, K=3}.
  v2f a;
  a[0] = (lane < 16) ? 1.0f : 0.0f;
  a[1] = 0.0f;
  // B (4x16): row K=0 (VGPR0, lanes 0-15) = bias segment; all other rows 0.
  v2f b;
  b[0] = (lane < 16) ? bias[col] : 0.0f;
  b[1] = 0.0f;
  c = __builtin_amdgcn_wmma_f32_16x16x4_f32(false, a, false, b, (short)0, c, false, false);
#else
  float bb = bias[col];
#pragma unroll
  for (int j = 0; j < 8; ++j) c[j] += bb;
#endif

#pragma unroll
  for (int j = 0; j < 8; ++j) {
    float v = c[j];
    v = v > 0.0f ? v : 0.0f;
    h[(size_t)(row0 + j) * UGCN_EMB + col] = v;
  }
}

// Scalar tail for N not divisible by 16 (empty for N=100000).
__global__ void ugcn_finalize_tail(float* __restrict__ h, const float* __restrict__ uw,
                                   const float* __restrict__ dinv,
                                   const float* __restrict__ bias, int start, int N) {
  int t = blockIdx.x * blockDim.x + threadIdx.x;
  int total = (N - start) * UGCN_EMB;
  if (t >= total) return;
  int row = start + (t >> 6);
  int col = t & 63;
  float di = dinv[row];
  size_t idx = (size_t)row * UGCN_EMB + col;
  float v = h[idx] + uw[idx] * (di * di) + bias[col];
  h[idx] = v > 0.0f ? v : 0.0f;
}

__global__ void ugcn_copy_items(const float4* __restrict__ src, float4* __restrict__ dst,
                                long long n4) {
  long long t = (long long)blockIdx.x * blockDim.x + threadIdx.x;
  if (t < n4) dst[t] = src[t];
}

extern "C" void kernel_launch(void* const* d_in, const int* in_sizes, int n_in,
                              void* d_out, int out_size, void* d_ws, size_t ws_size,
                              hipStream_t stream) {
  const int* edges = (const int*)d_in[0];    // (2, E) int32, row-major
  const float* uw = (const float*)d_in[1];   // (N, 64) f32
  const float* bias = (const float*)d_in[2]; // (64,) f32
  const float* items = (const float*)d_in[3];// (NI, 64) f32

  const int E = in_sizes[0] / 2;
  const int N = in_sizes[1] / UGCN_EMB;
  const int NI = in_sizes[3] / UGCN_EMB;

  const int* rows = edges;
  const int* cols = edges + E;

  float* out = (float*)d_out;     // [0, N*64) = user_h accumulator / result
  float* deg = (float*)d_ws;      // N floats
  float* dinv = deg + N;          // N floats

  const int B = 256;

  {  // init h = 0, deg = 1
    long long total = (long long)N * UGCN_EMB;
    int blocks = (int)((total + B - 1) / B);
    ugcn_init<<<blocks, B, 0, stream>>>(out, deg, N);
  }

  ugcn_degree<<<(E + B - 1) / B, B, 0, stream>>>(rows, deg, E);
  ugcn_rsqrt<<<(N + B - 1) / B, B, 0, stream>>>(deg, dinv, N);

  {  // edge scatter: 16 lanes per edge
    long long total = (long long)E * 16;
    int blocks = (int)((total + B - 1) / B);
    ugcn_scatter<<<blocks, B, 0, stream>>>(rows, cols, uw, dinv, out, E);
  }

  {  // finalize: WMMA tiles + scalar tail
    int tiles_m = N / 16;
    if (tiles_m > 0) {
      long long threads = (long long)tiles_m * 4 * 32;
      int blocks = (int)((threads + B - 1) / B);
      ugcn_finalize_wmma<<<blocks, B, 0, stream>>>(out, uw, dinv, bias, tiles_m);
    }
    int start = tiles_m * 16;
    if (start < N) {
      int total = (N - start) * UGCN_EMB;
      ugcn_finalize_tail<<<(total + B - 1) / B, B, 0, stream>>>(out, uw, dinv, bias, start, N);
    }
  }

  {  // item_embeddings passthrough into second half of d_out
    long long n4 = (long long)NI * (UGCN_EMB / 4);
    int blocks = (int)((n4 + B - 1) / B);
    ugcn_copy_items<<<blocks, B, 0, stream>>>((const float4*)items,
                                              (float4*)(out + (size_t)N * UGCN_EMB), n4);
  }
}